// SpecificMoE_54889682043310
// MI455X (gfx1250) — compile-verified
//
#include <hip/hip_runtime.h>
#include <hip/hip_bf16.h>
#include <math.h>

// Problem constants (fixed by the reference)
#define SEQ  4096
#define DIM  1024
#define NBAT 8
#define NEXP 8

typedef __attribute__((ext_vector_type(2))) float v2f;
typedef __attribute__((ext_vector_type(8))) float v8f;

static constexpr int BM  = 128;   // block tile M
static constexpr int BN  = 128;   // block tile N
static constexpr int KT  = 32;    // k-step staged in LDS
static constexpr int LDA = 36;    // padded LDS stride for A (16B rows, 16 distinct banks)
static constexpr int LDW = 132;   // padded LDS stride for W
static constexpr int NKB = DIM / KT;  // 32 k-blocks per GEMM

__device__ __forceinline__ v8f wmma4(v2f a, v2f b, v8f c) {
  // D = A(16x4 f32) * B(4x16 f32) + C
  return __builtin_amdgcn_wmma_f32_16x16x4_f32(false, a, false, b, (short)0, c, false, false);
}

__device__ __forceinline__ float gelu_exact(float v) {
  return 0.5f * v * (1.0f + erff(v * 0.70710678118654752f));
}

__device__ __forceinline__ uint32_t lds_addr32(const void* p) {
  // generic LDS pointer = {SHARED_BASE_hi, lds_offset[31:0]} -> low 32 bits are the DS address
  return (uint32_t)(uintptr_t)p;
}

__device__ __forceinline__ void async_b128(uint32_t lds_off, uint32_t glob_off,
                                           const float* base) {
  // GVS mode: mem = SADDR + VADDR(u32) + ioffset ; VDST = LDS byte address
  asm volatile("global_load_async_to_lds_b128 %0, %1, %2"
               :: "v"(lds_off), "v"(glob_off), "s"(base) : "memory");
}

__device__ __forceinline__ void wait_async0() {
  asm volatile("s_wait_asynccnt 0x0" ::: "memory");
}

// ---------------------------------------------------------------------------
// Router: scores0 = x[:,0,:] @ Wr + br  -> top-2 per batch (ties: lower idx)
// ---------------------------------------------------------------------------
__global__ __launch_bounds__(64)
void moe_router(const float* __restrict__ x, const float* __restrict__ Wr,
                const float* __restrict__ br, int* __restrict__ idx) {
  __shared__ float sc[NBAT][NEXP];
  const int t = threadIdx.x;
  if (t < NBAT * NEXP) {
    const int b = t / NEXP, e = t % NEXP;
    const float* xr = x + (size_t)b * SEQ * DIM;  // token 0
    float acc = 0.f;
    for (int i = 0; i < DIM; ++i) acc += xr[i] * Wr[i * NEXP + e];
    sc[b][e] = acc + br[e];
  }
  __syncthreads();
  if (t < NBAT) {
    int i0 = 0; float v0 = sc[t][0];
    for (int e = 1; e < NEXP; ++e) if (sc[t][e] > v0) { v0 = sc[t][e]; i0 = e; }
    int i1 = -1; float v1 = -3.4e38f;
    for (int e = 0; e < NEXP; ++e) if (e != i0 && sc[t][e] > v1) { v1 = sc[t][e]; i1 = e; }
    idx[t * 2 + 0] = i0;
    idx[t * 2 + 1] = i1;
  }
}

// ---------------------------------------------------------------------------
// GEMM machinery: 128x128 block tile, 8 waves (2x4), wave = 64x32 = 4x2 WMMA tiles
// ---------------------------------------------------------------------------
struct TileCtx {
  int rowBase, colBase, tid, lane, wm, wn, lrow, khalf;
};

__device__ __forceinline__ TileCtx make_ctx() {
  TileCtx c;
  c.rowBase = blockIdx.x * BM;
  c.colBase = blockIdx.y * BN;
  c.tid  = threadIdx.x;
  c.lane = c.tid & 31;
  const int w = c.tid >> 5;
  c.wm   = w & 1;
  c.wn   = w >> 1;
  c.lrow = c.lane & 15;
  c.khalf = (c.lane >> 4) << 1;   // lanes 0-15 -> K{0,1}, lanes 16-31 -> K{2,3}
  return c;
}

// Issue one k-block stage as async global->LDS DMA (8 x b128 per wave).
__device__ __forceinline__ void issue_stage(const TileCtx& c,
                                            const float* __restrict__ Ag,
                                            const float* __restrict__ Wg,
                                            int kb, uint32_t ldsA, uint32_t ldsW) {
  // A tile: 128 rows x 32 cols
  #pragma unroll
  for (int i = 0; i < 4; ++i) {
    const int f = c.tid + i * 256;
    const int r = f >> 3, c4 = (f & 7) << 2;
    const uint32_t go = (uint32_t)(((c.rowBase + r) * DIM + kb + c4) * 4);
    async_b128(ldsA + (uint32_t)((r * LDA + c4) * 4), go, Ag);
  }
  // W tile: 32 rows x 128 cols
  #pragma unroll
  for (int i = 0; i < 4; ++i) {
    const int f = c.tid + i * 256;
    const int r = f >> 5, c4 = (f & 31) << 2;
    const uint32_t go = (uint32_t)(((kb + r) * DIM + c.colBase + c4) * 4);
    async_b128(ldsW + (uint32_t)((r * LDW + c4) * 4), go, Wg);
  }
}

__device__ __forceinline__ void mma_kblock(const TileCtx& c,
                                           const float* __restrict__ sA,
                                           const float* __restrict__ sW,
                                           v8f acc[4][2]) {
  #pragma unroll
  for (int kk = 0; kk < KT; kk += 4) {
    v2f a[4], bfr[2];
    #pragma unroll
    for (int am = 0; am < 4; ++am) {
      const int r = c.wm * 64 + am * 16 + c.lrow;
      a[am] = *(const v2f*)&sA[r * LDA + kk + c.khalf];
    }
    #pragma unroll
    for (int bn = 0; bn < 2; ++bn) {
      const int col = c.wn * 32 + bn * 16 + c.lrow;
      v2f b;
      b.x = sW[(kk + c.khalf + 0) * LDW + col];
      b.y = sW[(kk + c.khalf + 1) * LDW + col];
      bfr[bn] = b;
    }
    #pragma unroll
    for (int am = 0; am < 4; ++am)
      #pragma unroll
      for (int bn = 0; bn < 2; ++bn)
        acc[am][bn] = wmma4(a[am], bfr[bn], acc[am][bn]);
  }
}

// ---------------------------------------------------------------------------
// GEMM1: h[slot] = GELU(x[b] @ W1[e] + b1[e]),  grid = (S/BM, D/BN, 2 slots)
// ---------------------------------------------------------------------------
__global__ __launch_bounds__(256)
void moe_gemm1(const float* __restrict__ x,  const float* __restrict__ W1,
               const float* __restrict__ b1, const int* __restrict__ idx,
               float* __restrict__ h, int batch) {
  __shared__ __align__(16) float sA[2][BM * LDA];
  __shared__ __align__(16) float sW[2][KT * LDW];

  const int slot = blockIdx.z;
  const int e    = idx[batch * 2 + slot];
  const float* Ag = x  + (size_t)batch * SEQ * DIM;
  const float* Wg = W1 + (size_t)e * DIM * DIM;
  const float* bg = b1 + (size_t)e * DIM;
  float*       Hg = h  + (size_t)slot * SEQ * DIM;

  const TileCtx c = make_ctx();
  const uint32_t la[2] = { lds_addr32(&sA[0][0]), lds_addr32(&sA[1][0]) };
  const uint32_t lw[2] = { lds_addr32(&sW[0][0]), lds_addr32(&sW[1][0]) };
  v8f acc[4][2] = {};

  issue_stage(c, Ag, Wg, 0, la[0], lw[0]);          // prologue: stage 0
  #pragma unroll 1
  for (int i = 0; i < NKB; ++i) {
    wait_async0();                                   // stage i landed in LDS
    __syncthreads();                                 // all waves done with buf i^1
    if (i + 1 < NKB)
      issue_stage(c, Ag, Wg, (i + 1) * KT, la[(i + 1) & 1], lw[(i + 1) & 1]);
    mma_kblock(c, sA[i & 1], sW[i & 1], acc);        // overlaps stage i+1 DMA
  }

  #pragma unroll
  for (int am = 0; am < 4; ++am)
    #pragma unroll
    for (int bn = 0; bn < 2; ++bn) {
      const int col   = c.colBase + c.wn * 32 + bn * 16 + c.lrow;
      const float bias = bg[col];
      const int rbase = c.rowBase + c.wm * 64 + am * 16 + ((c.lane >> 4) << 3);
      #pragma unroll
      for (int r = 0; r < 8; ++r) {
        float v = acc[am][bn][r] + bias;
        Hg[(size_t)(rbase + r) * DIM + col] = gelu_exact(v);
      }
    }
}

// ---------------------------------------------------------------------------
// GEMM2: out[b] = 0.5*(h0 @ W2[e0] + h1 @ W2[e1] + b2[e0] + b2[e1])
// Both slots folded into a single 64-stage pipelined K loop.
// ---------------------------------------------------------------------------
__global__ __launch_bounds__(256)
void moe_gemm2(const float* __restrict__ h,  const float* __restrict__ W2,
               const float* __restrict__ b2, const int* __restrict__ idx,
               float* __restrict__ out, int batch) {
  __shared__ __align__(16) float sA[2][BM * LDA];
  __shared__ __align__(16) float sW[2][KT * LDW];

  const int e0 = idx[batch * 2 + 0];
  const int e1 = idx[batch * 2 + 1];
  const float* Abase[2] = { h + (size_t)0 * SEQ * DIM, h + (size_t)1 * SEQ * DIM };
  const float* Wbase[2] = { W2 + (size_t)e0 * DIM * DIM, W2 + (size_t)e1 * DIM * DIM };

  const TileCtx c = make_ctx();
  const uint32_t la[2] = { lds_addr32(&sA[0][0]), lds_addr32(&sA[1][0]) };
  const uint32_t lw[2] = { lds_addr32(&sW[0][0]), lds_addr32(&sW[1][0]) };
  v8f acc[4][2] = {};

  issue_stage(c, Abase[0], Wbase[0], 0, la[0], lw[0]);
  #pragma unroll 1
  for (int t = 0; t < 2 * NKB; ++t) {
    wait_async0();
    __syncthreads();
    if (t + 1 < 2 * NKB) {
      const int ns = (t + 1) >= NKB;                 // slot of next stage
      const int nk = ((t + 1) - ns * NKB) * KT;
      issue_stage(c, Abase[ns], Wbase[ns], nk, la[(t + 1) & 1], lw[(t + 1) & 1]);
    }
    mma_kblock(c, sA[t & 1], sW[t & 1], acc);
  }

  float* Og = out + (size_t)batch * SEQ * DIM;
  #pragma unroll
  for (int am = 0; am < 4; ++am)
    #pragma unroll
    for (int bn = 0; bn < 2; ++bn) {
      const int col = c.colBase + c.wn * 32 + bn * 16 + c.lrow;
      const float bias = 0.5f * (b2[(size_t)e0 * DIM + col] + b2[(size_t)e1 * DIM + col]);
      const int rbase = c.rowBase + c.wm * 64 + am * 16 + ((c.lane >> 4) << 3);
      #pragma unroll
      for (int r = 0; r < 8; ++r)
        Og[(size_t)(rbase + r) * DIM + col] = 0.5f * acc[am][bn][r] + bias;
    }
}

// ---------------------------------------------------------------------------
extern "C" void kernel_launch(void* const* d_in, const int* in_sizes, int n_in,
                              void* d_out, int out_size, void* d_ws, size_t ws_size,
                              hipStream_t stream) {
  const float* x  = (const float*)d_in[0];
  const float* Wr = (const float*)d_in[1];
  const float* br = (const float*)d_in[2];
  const float* W1 = (const float*)d_in[3];
  const float* b1 = (const float*)d_in[4];
  const float* W2 = (const float*)d_in[5];
  const float* b2 = (const float*)d_in[6];
  float* out = (float*)d_out;

  // workspace: h[2][SEQ][DIM] f32 (32 MB), then idx[NBAT][2] int
  float* h = (float*)d_ws;
  int* idx = (int*)((char*)d_ws + (size_t)2 * SEQ * DIM * sizeof(float));

  moe_router<<<1, 64, 0, stream>>>(x, Wr, br, idx);

  for (int b = 0; b < NBAT; ++b) {
    moe_gemm1<<<dim3(SEQ / BM, DIM / BN, 2), 256, 0, stream>>>(x, W1, b1, idx, h, b);
    moe_gemm2<<<dim3(SEQ / BM, DIM / BN),    256, 0, stream>>>(h, W2, b2, idx, out, b);
  }
}